// RBF_65094524338658
// MI455X (gfx1250) — compile-verified
//
#include <hip/hip_runtime.h>
#include <hip/hip_bf16.h>

typedef __attribute__((ext_vector_type(2))) float v2f;
typedef __attribute__((ext_vector_type(8))) float v8f;

#define NROWS 8192
#define MROWS 8192
#define DDIM  64

// stable softplus: logaddexp(x, 0) = max(x,0) + log1p(exp(-|x|))
__device__ __forceinline__ float softplus_f(float x) {
    return fmaxf(x, 0.0f) + log1pf(__expf(-fabsf(x)));
}

// ---------------------------------------------------------------------------
// Kernel 1: Xl = X / softplus(ls), X2l = X2 / softplus(ls), row norms xn, yn.
// One row per thread (tiny 4 MB prologue, L2 absorbs it).
// ---------------------------------------------------------------------------
__global__ __launch_bounds__(256) void rbf_prep_kernel(
    const float* __restrict__ X, const float* __restrict__ X2,
    const float* __restrict__ ls,
    float* __restrict__ Xl, float* __restrict__ X2l,
    float* __restrict__ xn, float* __restrict__ yn) {
    int row = blockIdx.x * blockDim.x + threadIdx.x;   // 0 .. N+M-1
    const float* src;
    float* dst;
    float* nrm;
    int r;
    if (row < NROWS) { src = X;  dst = Xl;  nrm = xn; r = row; }
    else             { src = X2; dst = X2l; nrm = yn; r = row - NROWS; }
    float s = 0.0f;
#pragma unroll 8
    for (int d = 0; d < DDIM; ++d) {
        float l = softplus_f(ls[d]);
        float v = src[r * DDIM + d] / l;
        dst[r * DDIM + d] = v;
        s += v * v;
    }
    nrm[r] = s;
}

// ---------------------------------------------------------------------------
// Kernel 2: cross = Xl @ X2l^T via V_WMMA_F32_16X16X4_F32, fused RBF epilogue.
// Each wave computes a 16x64 output strip (4 16x16 tiles sharing the A frag).
// 256 threads = 8 waves per block; 512 tile-rows x 128 col-groups = 65536
// wave tasks = 8192 blocks. EXEC is all-ones everywhere (WMMA requirement).
// ---------------------------------------------------------------------------
__global__ __launch_bounds__(256) void rbf_wmma_kernel(
    const float* __restrict__ Xl, const float* __restrict__ X2l,
    const float* __restrict__ xn, const float* __restrict__ yn,
    const float* __restrict__ variance, float* __restrict__ out) {

    const int lane = threadIdx.x & 31;
    const int wave = threadIdx.x >> 5;
    const int task = blockIdx.x * 8 + wave;     // 0 .. 65535
    const int tm   = task >> 7;                 // tile row   (16 rows)
    const int tg   = task & 127;                // col group  (64 cols)
    const int row0 = tm * 16;
    const int col0 = tg * 64;

    const int half = lane >> 4;                 // 0 | 1
    const int l16  = lane & 15;

    // A frag (16x4 f32): lane l16 -> M=l16 ; VGPR v -> K = 4*s + 2*half + v
    const float* aptr = Xl  + (size_t)(row0 + l16) * DDIM + 2 * half;
    // B frag (4x16 f32): lane l16 -> N=l16 ; same K mapping; B[k][n]=X2l[col][k]
    const float* bptr = X2l + (size_t)(col0 + l16) * DDIM + 2 * half;

    v8f c0 = {}, c1 = {}, c2 = {}, c3 = {};

#pragma unroll
    for (int s = 0; s < 16; ++s) {
        v2f a  = *(const v2f*)(aptr + 4 * s);
        v2f b0 = *(const v2f*)(bptr +  0 * DDIM + 4 * s);
        v2f b1 = *(const v2f*)(bptr + 16 * DDIM + 4 * s);
        v2f b2 = *(const v2f*)(bptr + 32 * DDIM + 4 * s);
        v2f b3 = *(const v2f*)(bptr + 48 * DDIM + 4 * s);
        // 8 args: (neg_a, A, neg_b, B, c_mod, C, reuse_a, reuse_b)
        c0 = __builtin_amdgcn_wmma_f32_16x16x4_f32(false, a, false, b0, (short)0, c0, false, false);
        c1 = __builtin_amdgcn_wmma_f32_16x16x4_f32(false, a, false, b1, (short)0, c1, false, false);
        c2 = __builtin_amdgcn_wmma_f32_16x16x4_f32(false, a, false, b2, (short)0, c2, false, false);
        c3 = __builtin_amdgcn_wmma_f32_16x16x4_f32(false, a, false, b3, (short)0, c3, false, false);
    }

    // Fused epilogue: d2 = max(xn + yn - 2*cross, 0); out = v * exp(-0.5*d2)
    const float vscale = softplus_f(variance[0]);

    float ynv0 = yn[col0 +  0 + l16];
    float ynv1 = yn[col0 + 16 + l16];
    float ynv2 = yn[col0 + 32 + l16];
    float ynv3 = yn[col0 + 48 + l16];

#pragma unroll
    for (int i = 0; i < 8; ++i) {
        // D layout: VGPR i -> row = i + 8*half, col = l16 (+16*j per tile)
        const int r = row0 + i + 8 * half;
        const float xv = xn[r];
        float* orow = out + (size_t)r * MROWS + col0 + l16;

        float d0 = fmaxf(xv + ynv0 - 2.0f * c0[i], 0.0f);
        float d1 = fmaxf(xv + ynv1 - 2.0f * c1[i], 0.0f);
        float d2 = fmaxf(xv + ynv2 - 2.0f * c2[i], 0.0f);
        float d3 = fmaxf(xv + ynv3 - 2.0f * c3[i], 0.0f);

        orow[ 0] = vscale * __expf(-0.5f * d0);
        orow[16] = vscale * __expf(-0.5f * d1);
        orow[32] = vscale * __expf(-0.5f * d2);
        orow[48] = vscale * __expf(-0.5f * d3);
    }
}

// ---------------------------------------------------------------------------
extern "C" void kernel_launch(void* const* d_in, const int* in_sizes, int n_in,
                              void* d_out, int out_size, void* d_ws, size_t ws_size,
                              hipStream_t stream) {
    const float* X   = (const float*)d_in[0];   // [8192, 64]
    const float* X2  = (const float*)d_in[1];   // [8192, 64]
    const float* ls  = (const float*)d_in[2];   // [1, 64]
    const float* var = (const float*)d_in[3];   // [1]
    float* out = (float*)d_out;                 // [8192, 8192]

    float* ws  = (float*)d_ws;
    float* Xl  = ws;                            // 8192*64
    float* X2l = Xl  + (size_t)NROWS * DDIM;    // 8192*64
    float* xn  = X2l + (size_t)MROWS * DDIM;    // 8192
    float* yn  = xn  + NROWS;                   // 8192

    rbf_prep_kernel<<<(NROWS + MROWS) / 256, 256, 0, stream>>>(X, X2, ls, Xl, X2l, xn, yn);

    // 512 tile-rows * 128 col-groups = 65536 wave tasks / 8 waves per block
    rbf_wmma_kernel<<<65536 / 8, 256, 0, stream>>>(Xl, X2l, xn, yn, var, out);
}